// ANDNLayer_56538949485245
// MI455X (gfx1250) — compile-verified
//
#include <hip/hip_runtime.h>

// Problem constants (fixed by the reference).
#define B_ 64
#define N_ 65536
#define D_ 32768
#define K_ 16
#define NEG_INF_REF (-1e30f)

#ifndef __has_builtin
#define __has_builtin(x) 0
#endif

// gfx1250 async global->LDS copy path (ASYNCcnt-tracked). Probe-confirmed to
// exist on this toolchain; signature is (v4i AS1*, v4i AS3*, imm, imm).
#if __has_builtin(__builtin_amdgcn_global_load_async_to_lds_b128) && \
    __has_builtin(__builtin_amdgcn_s_wait_asynccnt)
#define USE_ASYNC_LDS 1
#else
#define USE_ASYNC_LDS 0
#endif

typedef int v4i_ __attribute__((vector_size(4 * sizeof(int))));
typedef v4i_ __attribute__((address_space(1))) gbl_v4i;
typedef v4i_ __attribute__((address_space(3))) lds_v4i;

// ---------------------------------------------------------------------------
// Kernel: stage x[b,:] (256 KB) into LDS via async-to-LDS, then per-detector
// top-2 gather, winner/prewinner ids out, per-batch winner histogram via
// global atomics. gridDim = (tilesPerBatch, B), blockDim = 1024,
// dynamic LDS = N*4 bytes (256 KB of the WGP's 320 KB).
// Defined FIRST in the file so the disasm snippet shows its staging prologue.
// ---------------------------------------------------------------------------
__global__ __launch_bounds__(1024, 1) void k_top2_winner(
    const float* __restrict__ x, const int* __restrict__ det,
    float* __restrict__ win_out, float* __restrict__ prewin_out,
    int* __restrict__ wcount) {
  extern __shared__ float xrow[];  // N_ floats = 256 KB

  const int b    = blockIdx.y;
  const int tile = blockIdx.x;
  const int tid  = threadIdx.x;
  const int nthr = blockDim.x;

  const float* xr = x + (size_t)b * N_;

  // ---- Stage the whole row into LDS, 16 B per lane per iteration ----------
  for (int off = tid * 4; off < N_; off += nthr * 4) {
#if USE_ASYNC_LDS
    __builtin_amdgcn_global_load_async_to_lds_b128(
        (gbl_v4i*)(xr + off), (lds_v4i*)(xrow + off),
        /*offset=*/0, /*cpol=*/0);
#else
    const float4 v = *(const float4*)(xr + off);
    *(float4*)(xrow + off) = v;
#endif
  }
#if USE_ASYNC_LDS
  __builtin_amdgcn_s_wait_asynccnt(0);
#endif
  __syncthreads();

  // ---- Top-2 over K=16 gathered values per detector -----------------------
  const int chunk   = D_ / gridDim.x;
  const int d_begin = tile * chunk;
  const int d_end   = d_begin + chunk;

  for (int d = d_begin + tid; d < d_end; d += nthr) {
    const int4* ip = (const int4*)(det + (size_t)d * K_);
    // Prefetch the next detector block we will touch (speculative is fine).
    __builtin_prefetch(det + (size_t)(d + nthr) * K_, 0, 0);

    int4 q0 = ip[0], q1 = ip[1], q2 = ip[2], q3 = ip[3];

    float best = -__builtin_inff();
    int best_id = 0;
    float sec = -__builtin_inff();
    int sec_id = 0;

    auto consider = [&](int id) {
      const bool valid = id >= 0;
      const int cid = valid ? id : 0;
      const float v = valid ? xrow[cid] : NEG_INF_REF;  // ds_load_b32 gather
      if (v > best) {          // strict '>' => stable (earlier index wins)
        sec = best; sec_id = best_id;
        best = v;  best_id = cid;
      } else if (v > sec) {
        sec = v; sec_id = cid;
      }
    };
    consider(q0.x); consider(q0.y); consider(q0.z); consider(q0.w);
    consider(q1.x); consider(q1.y); consider(q1.z); consider(q1.w);
    consider(q2.x); consider(q2.y); consider(q2.z); consider(q2.w);
    consider(q3.x); consider(q3.y); consider(q3.z); consider(q3.w);

    const size_t o = (size_t)b * D_ + d;
    win_out[o]    = (float)best_id;   // d_out is a flat float32 buffer
    prewin_out[o] = (float)sec_id;
    atomicAdd(&wcount[(size_t)b * N_ + best_id], 1);
  }
}

// ---------------------------------------------------------------------------
// Kernel: member_count histogram over all detector indices (batch-invariant)
// ---------------------------------------------------------------------------
__global__ void k_member_hist(const int* __restrict__ det,
                              int* __restrict__ member, int total) {
  int i = blockIdx.x * blockDim.x + threadIdx.x;
  if (i < total) {
    int id = det[i];
    if (id >= 0) atomicAdd(&member[id], 1);
  }
}

// ---------------------------------------------------------------------------
// Kernel: output = (member - winner_count == 0), stat = member - winner.
// 4-wide vectorized; N is a power of two so n = i & (N-1).
// ---------------------------------------------------------------------------
__global__ void k_finalize(const int* __restrict__ member,
                           const int* __restrict__ wcount,
                           float* __restrict__ out,
                           float* __restrict__ stat) {
  const size_t gid = (size_t)blockIdx.x * blockDim.x + threadIdx.x;
  const size_t i4 = gid * 4;  // total B_*N_ elements, exact grid
  const int n0 = (int)(i4 & (size_t)(N_ - 1));

  int4 m = *(const int4*)(member + n0);
  int4 w = *(const int4*)(wcount + i4);
  int4 s = make_int4(m.x - w.x, m.y - w.y, m.z - w.z, m.w - w.w);

  float4 o = make_float4(s.x == 0 ? 1.f : 0.f, s.y == 0 ? 1.f : 0.f,
                         s.z == 0 ? 1.f : 0.f, s.w == 0 ? 1.f : 0.f);
  float4 sf = make_float4((float)s.x, (float)s.y, (float)s.z, (float)s.w);

  *(float4*)(out + i4) = o;
  *(float4*)(stat + i4) = sf;
}

// ---------------------------------------------------------------------------
extern "C" void kernel_launch(void* const* d_in, const int* in_sizes, int n_in,
                              void* d_out, int out_size, void* d_ws,
                              size_t ws_size, hipStream_t stream) {
  (void)in_sizes; (void)n_in; (void)out_size; (void)ws_size;

  const float* x  = (const float*)d_in[0];
  const int* det  = (const int*)d_in[1];

  // d_out layout (flat float32, reference tuple order):
  //   [0, B*N)                       output
  //   [B*N, B*N + B*D)               winner_ids
  //   [B*N + B*D, B*N + 2*B*D)       prewinner_ids
  //   [B*N + 2*B*D, 2*B*N + 2*B*D)   input_winning_stat
  float* out_output = (float*)d_out;
  float* out_winner = out_output + (size_t)B_ * N_;
  float* out_prewin = out_winner + (size_t)B_ * D_;
  float* out_stat   = out_prewin + (size_t)B_ * D_;

  // Workspace: member_count (N ints) then winner_count (B*N ints) = 16.25 MB.
  int* member = (int*)d_ws;
  int* wcount = member + N_;
  (void)hipMemsetAsync(d_ws, 0,
                       ((size_t)N_ + (size_t)B_ * N_) * sizeof(int), stream);

  // Phase 1: member histogram (512K updates).
  {
    const int total = D_ * K_;
    k_member_hist<<<(total + 1023) / 1024, 1024, 0, stream>>>(det, member,
                                                              total);
  }

  // Phase 2: LDS-staged top-2 + winner histogram.
  {
    const int tilesPerBatch = 8;             // 512 WGs, 1 per WGP (256KB LDS)
    const size_t lds = (size_t)N_ * sizeof(float);
    (void)hipFuncSetAttribute(
        (const void*)k_top2_winner,
        hipFuncAttributeMaxDynamicSharedMemorySize, (int)lds);
    dim3 grid(tilesPerBatch, B_);
    k_top2_winner<<<grid, 1024, lds, stream>>>(x, det, out_winner, out_prewin,
                                               wcount);
  }

  // Phase 3: finalize (vectorized x4, exact grid: B*N/4 / 256 blocks).
  {
    const size_t total4 = (size_t)B_ * N_ / 4;
    k_finalize<<<(unsigned)(total4 / 256), 256, 0, stream>>>(
        member, wcount, out_output, out_stat);
  }
}